// ConvPolicy8_20761871909589
// MI455X (gfx1250) — compile-verified
//
#include <hip/hip_runtime.h>

typedef __attribute__((ext_vector_type(2))) float v2f;
typedef __attribute__((ext_vector_type(8))) float v8f;

// tanh(x) = 1 - 2/(e^{2x}+1): v_exp_f32 + v_rcp_f32 + 3 VALU, ~1e-6 rel err.
// Saturates correctly: x->+inf => 1, x->-inf => -1, NaN propagates.
__device__ __forceinline__ float fast_tanh(float x) {
  float e = __expf(2.0f * x);
  return 1.0f - 2.0f * __builtin_amdgcn_rcpf(e + 1.0f);
}

// Weight element, output row mc (pre-clamped to <O), reduction index rr
// (compile-time after unroll). Conv w(O,I,KW): w[m*I*KW + r].
// Deconv w(I,O,KW) flipped: w[i*O*KW + m*KW + (KW-1-k)].
// Both are  (m * const_stride) + const_offset  => global_load with imm offset.
template<int O, int I, int KW, bool DC>
__device__ __forceinline__ float wld(const float* __restrict__ w, int mc, int rr) {
  if (rr >= I * KW) return 0.0f;                 // tail K slots: constexpr zero
  int i = rr / KW, k = rr - i * KW;              // folds at compile time
  int idx = DC ? (i * (O * KW) + mc * KW + (KW - 1 - k))
               : (mc * (I * KW) + rr);
  return w[idx];
}

// One layer as a chain of V_WMMA_F32_16X16X4_F32.
// D[M=out_ch, N=out_pos]; A = weights (16x4 chunk), B = prebuilt im2col rows.
// Lane-half specialization makes all offsets compile-time immediates.
template<int O, int I, int KW, int LOUT, bool DC>
__device__ __forceinline__ v8f mm(const float* __restrict__ w,
                                  const float* im, int lane) {
  constexpr int KT = I * KW;
  constexpr int NC = (KT + 3) / 4;
  v2f A[NC], B[NC];
  const int n = lane & 15;
  if (lane < 16) {                               // half 0: K slots 4c+0, 4c+1
    const int mc = (n < O) ? n : (O - 1);        // clamp: rows m>=O read dup row
#pragma unroll
    for (int c = 0; c < NC; ++c) {
      A[c].x = wld<O, I, KW, DC>(w, mc, 4 * c + 0);
      A[c].y = wld<O, I, KW, DC>(w, mc, 4 * c + 1);
      B[c].x = im[(4 * c + 0) * LOUT + n];       // ds_load, imm offset
      B[c].y = im[(4 * c + 1) * LOUT + n];
    }
  } else {                                       // half 1: K slots 4c+2, 4c+3
    const int mc = (n < O) ? n : (O - 1);
#pragma unroll
    for (int c = 0; c < NC; ++c) {
      A[c].x = wld<O, I, KW, DC>(w, mc, 4 * c + 2);
      A[c].y = wld<O, I, KW, DC>(w, mc, 4 * c + 3);
      B[c].x = im[(4 * c + 2) * LOUT + n];
      B[c].y = im[(4 * c + 3) * LOUT + n];
    }
  }
  v8f acc = {};                                  // EXEC fully reconverged here
#pragma unroll
  for (int c = 0; c < NC; ++c)
    acc = __builtin_amdgcn_wmma_f32_16x16x4_f32(false, A[c], false, B[c],
                                                (short)0, acc, false, false);
  return acc;
}

// Zero NF floats of an LDS buffer with b128 stores (NF % 4 == 0).
template<int NF>
__device__ __forceinline__ void zfill(float* p, int lane) {
  constexpr int N4 = NF / 4;
#pragma unroll
  for (int b = 0; b < N4; b += 32)
    if (b + lane < N4)
      reinterpret_cast<float4*>(p)[b + lane] = make_float4(0.f, 0.f, 0.f, 0.f);
}

// Generic epilogue (lanes 0-15 only): bias + optional tanh, then scatter the
// value (row r, position n) into the NEXT layer's im2col at rows r*KWn+k,
// column n + PADn - k. Cells not covered stay zero (pre-zeroed buffer).
template<int O, int LOUT, bool ACT, int KWn, int PADn, int LOUTn>
__device__ __forceinline__ void epi(const v8f& acc, const float* __restrict__ bias,
                                    float* im, int n) {
  float v[O];
#pragma unroll
  for (int r = 0; r < O; ++r) {
    float t = acc[r] + bias[r];                  // bias[r]: uniform s_load
    v[r] = ACT ? fast_tanh(t) : t;
  }
#pragma unroll
  for (int k = 0; k < KWn; ++k) {
    const int lo = (k > PADn) ? (k - PADn) : 0;  // need 0 <= n+PADn-k < LOUTn
    const int h0 = LOUTn + k - PADn;
    const int hi = (h0 < LOUT) ? h0 : LOUT;      // and n < LOUT (valid source)
    if (hi > lo && (unsigned)(n - lo) < (unsigned)(hi - lo)) {
#pragma unroll
      for (int r = 0; r < O; ++r)
        im[(r * KWn + k) * LOUTn + (n + PADn - k)] = v[r];
    }
  }
}

// d3 epilogue: fold fm_up = fm_dc3[:, :, [0,0,1,2]] into the d4 im2col scatter.
// d3 value (row r, pos n) feeds fm_up positions p with posmap[p]==n:
//   n=0 -> p in {0,1}; n=1 -> p=2; n=2 -> p=3.  Column nn = p + 1 - k.
__device__ __forceinline__ void epiD3(const v8f& acc, const float* __restrict__ bias,
                                      float* im, int n) {
  float v[8];
#pragma unroll
  for (int r = 0; r < 8; ++r) v[r] = fast_tanh(acc[r] + bias[r]);
  if (n == 0) {                                  // duplicate target p = 1
#pragma unroll
    for (int k = 0; k < 3; ++k)
#pragma unroll
      for (int r = 0; r < 8; ++r)
        im[(r * 3 + k) * 4 + (2 - k)] = v[r];
  }
  int l = (n == 0) ? 0 : (n + 1);                // main target p in {0,2,3}
#pragma unroll
  for (int k = 0; k < 3; ++k) {
    int nn = l + 1 - k;
    if (n < 3 && (unsigned)nn < 4u) {
#pragma unroll
      for (int r = 0; r < 8; ++r)
        im[(r * 3 + k) * 4 + nn] = v[r];
    }
  }
}

// d4 epilogue: bias only (no tanh); acts.reshape(24)[2:] -> out[22]
__device__ __forceinline__ void epiOut(const v8f& acc, const float* __restrict__ bias,
                                       float* __restrict__ out, int n) {
  if (n < 4) {
#pragma unroll
    for (int r = 0; r < 6; ++r) {
      float v = acc[r] + bias[r];
      if (r > 0 || n >= 2)
        out[r * 4 + n - 2] = v;
    }
  }
}

__global__ void __launch_bounds__(32)
conv_policy8_kernel(const float* __restrict__ x,
                    const float* __restrict__ wc1, const float* __restrict__ bc1,
                    const float* __restrict__ wc2, const float* __restrict__ bc2,
                    const float* __restrict__ wc3, const float* __restrict__ bc3,
                    const float* __restrict__ wc4, const float* __restrict__ bc4,
                    const float* __restrict__ we1, const float* __restrict__ be1,
                    const float* __restrict__ we2, const float* __restrict__ be2,
                    const float* __restrict__ wd1, const float* __restrict__ bd1,
                    const float* __restrict__ wd2, const float* __restrict__ bd2,
                    const float* __restrict__ wd3, const float* __restrict__ bd3,
                    const float* __restrict__ wd4, const float* __restrict__ bd4,
                    float* __restrict__ out) {
  __shared__ __align__(16) float bufc1[144];  // c1 im2col: 36 rows x 4
  __shared__ __align__(16) float bufA[80];    // ping (max 72: d1 input)
  __shared__ __align__(16) float bufB[80];    // pong
  __shared__ __align__(16) float buf_e[12];   // e1 im2col: 12 rows x 1
  __shared__ __align__(16) float buf3[176];   // d4 im2col: 44 rows x 4
  const int lane = threadIdx.x;
  const int n = lane & 15;

  // ---- one-time static builds (overlap first layer's weight latency) ----
  // c1 im2col from x: row r=i*3+k (i<12, k<3), col n0<4, value jcat[i][n0-1+k].
  // jcat flat[0..47] = [0,0,x[7..28], 0,0,x[35..56]]
#pragma unroll
  for (int b = 0; b < 160; b += 32) {
    int e = b + lane;
    if (e < 144) {
      int r = e >> 2, n0 = e & 3;
      int i = r / 3, k = r - 3 * i;
      int p = n0 - 1 + k;
      int q = i * 4 + p;
      int t = (q < 24) ? q : (q - 24);
      bool ok = (p >= 0) && (p < 4) && (t >= 2);
      int xi = (q < 24) ? (5 + q) : (9 + q);    // x[5+q] / x[33+(q-24)]
      float xv = x[ok ? xi : 0];
      bufc1[e] = ok ? xv : 0.0f;
    }
  }
  // d4 im2col: rows 0..23 (fm_up region) zero; rows 24..41 = jlrs; 42..43 zero
  zfill<96>(buf3, lane);
#pragma unroll
  for (int b = 0; b < 96; b += 32) {
    int e2 = b + lane;
    if (e2 < 80) {
      int e = 96 + e2;
      int r = (e >> 2) - 24;                    // 0..19
      int n0 = e & 3;
      float val = 0.0f;
      if (r < 18) {
        int i6 = r / 3, k = r - 3 * i6;
        int p = n0 - 1 + k;
        int q = i6 * 4 + p;                     // jlrs flat index
        bool ok = (p >= 0) && (p < 4) && (q >= 2);
        float xv = x[ok ? (5 + q) : 0];
        val = ok ? xv : 0.0f;
      }
      buf3[e] = val;
    }
  }
  // e1 im2col static rows: 8=psi, 9=x[34], 10..11 = K tail zeros
  if (lane == 0) {
    float qw = x[3], qx = x[4], qy = x[5], qz = x[6];
    buf_e[8]  = atan2f(qz, qw) - atan2f(-qx, qy);
    buf_e[9]  = x[34];
    buf_e[10] = 0.0f;
    buf_e[11] = 0.0f;
  }

  // ---- layer chain (single wave; LDS ops are in-order within a wave) ----
  v8f a;
  a = mm<4, 12, 3, 4, false>(wc1, bufc1, lane);            // fm_c1
  zfill<48>(bufA, lane);
  if (lane < 16) epi<4, 4, true, 3, 1, 4>(a, bc1, bufA, n);

  a = mm<8, 4, 3, 4, false>(wc2, bufA, lane);              // fm_c2
  zfill<48>(bufB, lane);
  if (lane < 16) epi<8, 4, true, 3, 0, 2>(a, bc2, bufB, n);

  a = mm<8, 8, 3, 2, false>(wc3, bufB, lane);              // fm_c3
  zfill<16>(bufA, lane);
  if (lane < 16) epi<8, 2, true, 2, 0, 1>(a, bc3, bufA, n);

  a = mm<8, 8, 2, 1, false>(wc4, bufA, lane);              // fm_c4
  if (lane < 16) epi<8, 1, true, 1, 0, 1>(a, bc4, buf_e, n); // writes rows 0..7

  a = mm<8, 10, 1, 1, false>(we1, buf_e, lane);            // emb_1
  if (lane < 16) epi<8, 1, true, 1, 0, 1>(a, be1, bufB, n);  // all 8 cells

  a = mm<8, 8, 1, 1, false>(we2, bufB, lane);              // emb_2
  zfill<72>(bufA, lane);
  if (lane < 16) epi<8, 1, true, 3, 2, 3>(a, be2, bufA, n);

  a = mm<4, 8, 3, 3, true>(wd1, bufA, lane);               // fm_dc1 (deconv)
  zfill<36>(bufB, lane);
  if (lane < 16) epi<4, 3, true, 3, 1, 3>(a, bd1, bufB, n);

  a = mm<4, 4, 3, 3, true>(wd2, bufB, lane);               // fm_dc2
  zfill<36>(bufA, lane);
  if (lane < 16) epi<4, 3, true, 3, 1, 3>(a, bd2, bufA, n);

  a = mm<8, 4, 3, 3, true>(wd3, bufA, lane);               // fm_dc3
  if (lane < 16) epiD3(a, bd3, buf3, n);                   // + [0,0,1,2] gather

  a = mm<6, 14, 3, 4, true>(wd4, buf3, lane);              // fm_dc4
  if (lane < 16) epiOut(a, bd4, out, n);
}

extern "C" void kernel_launch(void* const* d_in, const int* in_sizes, int n_in,
                              void* d_out, int out_size, void* d_ws, size_t ws_size,
                              hipStream_t stream) {
  (void)in_sizes; (void)n_in; (void)out_size; (void)d_ws; (void)ws_size;
  const float* x   = (const float*)d_in[0];
  const float* wc1 = (const float*)d_in[1];  const float* bc1 = (const float*)d_in[2];
  const float* wc2 = (const float*)d_in[3];  const float* bc2 = (const float*)d_in[4];
  const float* wc3 = (const float*)d_in[5];  const float* bc3 = (const float*)d_in[6];
  const float* wc4 = (const float*)d_in[7];  const float* bc4 = (const float*)d_in[8];
  const float* we1 = (const float*)d_in[9];  const float* be1 = (const float*)d_in[10];
  const float* we2 = (const float*)d_in[11]; const float* be2 = (const float*)d_in[12];
  const float* wd1 = (const float*)d_in[13]; const float* bd1 = (const float*)d_in[14];
  const float* wd2 = (const float*)d_in[15]; const float* bd2 = (const float*)d_in[16];
  const float* wd3 = (const float*)d_in[17]; const float* bd3 = (const float*)d_in[18];
  const float* wd4 = (const float*)d_in[19]; const float* bd4 = (const float*)d_in[20];

  conv_policy8_kernel<<<1, 32, 0, stream>>>(x,
      wc1, bc1, wc2, bc2, wc3, bc3, wc4, bc4,
      we1, be1, we2, be2,
      wd1, bd1, wd2, bd2, wd3, bd3, wd4, bd4,
      (float*)d_out);
}